// MultiHeadedAttention_2903397892960
// MI455X (gfx1250) — compile-verified
//
#include <hip/hip_runtime.h>
#include <math.h>
#include <stdint.h>

// ---------------------------------------------------------------------------
// MultiHeadedAttention forward, fp32, CDNA5 (gfx1250) WMMA path.
// B=2, T=2048, D=1024, H=16, HD=64
// Pipeline: [x @ w_qkv] -> [causal flash attention] -> [vals @ w_out]
// All matmuls on v_wmma_f32_16x16x4_f32 (full fp32 fidelity).
// Global->LDS staging via GLOBAL_LOAD_ASYNC_TO_LDS_B128 (ASYNCcnt),
// double-buffered so slab k+1 streams in while slab k is computed.
// ---------------------------------------------------------------------------

typedef __attribute__((ext_vector_type(2))) float v2f;
typedef __attribute__((ext_vector_type(8))) float v8f;

#define BB  2
#define T_  2048
#define D_  1024
#define H_  16
#define HD_ 64
#define D3  3072   // 3*D

// D = A(16x4 f32) * B(4x16 f32) + C(16x16 f32)  -> v_wmma_f32_16x16x4_f32
__device__ __forceinline__ v8f wmma_f32(v2f a, v2f b, v8f c) {
  return __builtin_amdgcn_wmma_f32_16x16x4_f32(
      /*neg_a=*/false, a, /*neg_b=*/false, b,
      /*c_mod=*/(short)0, c, /*reuse_a=*/false, /*reuse_b=*/false);
}

// Async 16B global -> LDS copy (per lane), tracked by ASYNCcnt.
// LDS address = low 32 bits of the flat pointer.
__device__ __forceinline__ void async_copy_b128(void* lds_ptr, const void* gptr) {
  uint32_t lds_off = (uint32_t)(uintptr_t)lds_ptr;
  asm volatile("global_load_async_to_lds_b128 %0, %1, off"
               :: "v"(lds_off), "v"(gptr)
               : "memory");
}
__device__ __forceinline__ void wait_asynccnt0() {
  asm volatile("s_wait_asynccnt 0" ::: "memory");
}

// ---------------------------------------------------------------------------
// Generic row-major fp32 GEMM: C[M,N] = A[M,K] * B[K,N]
// Block tile 128x128, 256 threads (8 waves in 2x4 grid, 64x32 per wave).
// K staged in 32-wide slabs, double-buffered via async-to-LDS.
// M,N multiples of 128; K multiple of 32.
// ---------------------------------------------------------------------------
#define AS_STRIDE 36   // 32+4 pad: rows 144B (16B aligned), distinct banks

__global__ __launch_bounds__(256)
void gemm_f32_wmma_k(const float* __restrict__ A, const float* __restrict__ B,
                     float* __restrict__ C, int M, int N, int K)
{
  (void)M;
  extern __shared__ float smem[];
  float* AsB = smem;                         // [2][128*AS_STRIDE]
  float* BsB = smem + 2 * 128 * AS_STRIDE;   // [2][32*128]

  const int tid  = threadIdx.x;
  const int lane = tid & 31;
  const int w    = tid >> 5;      // wave 0..7
  const int wm   = w >> 2;        // 0..1  (M dir)
  const int wn   = w & 3;         // 0..3  (N dir)
  const int l16  = lane & 15;
  const int kh   = lane >> 4;     // half-wave -> K pair select
  const int m0   = blockIdx.y * 128;
  const int n0   = blockIdx.x * 128;

  v8f acc[4][2];
  #pragma unroll
  for (int mt = 0; mt < 4; ++mt)
    #pragma unroll
    for (int nt = 0; nt < 2; ++nt)
      #pragma unroll
      for (int v = 0; v < 8; ++v)
        acc[mt][nt][v] = 0.0f;

  // ---- async stage of one K-slab (A: 128x32, B: 32x128) into buffer ib ----
  auto stage = [&](int k0, int ib) {
    float* As = AsB + ib * (128 * AS_STRIDE);
    float* Bs = BsB + ib * (32 * 128);
    #pragma unroll
    for (int i = 0; i < 4; ++i) {
      int idx = tid + i * 256;            // 128 rows x 8 float4 slots
      int row = idx >> 3, c4 = idx & 7;
      async_copy_b128(&As[row * AS_STRIDE + c4 * 4],
                      A + (size_t)(m0 + row) * K + k0 + c4 * 4);
    }
    #pragma unroll
    for (int i = 0; i < 4; ++i) {
      int idx = tid + i * 256;            // 32 rows x 32 float4 slots
      int row = idx >> 5, c4 = idx & 31;
      async_copy_b128(&Bs[row * 128 + c4 * 4],
                      B + (size_t)(k0 + row) * N + n0 + c4 * 4);
    }
  };

  stage(0, 0);
  int ib = 0;
  for (int k0 = 0; k0 < K; k0 += 32, ib ^= 1) {
    wait_asynccnt0();      // my buffer-ib copies done
    __syncthreads();       // everyone's done; prior readers of ib^1 finished
    if (k0 + 32 < K) stage(k0 + 32, ib ^ 1);

    const float* As = AsB + ib * (128 * AS_STRIDE);
    const float* Bs = BsB + ib * (32 * 128);

    #pragma unroll
    for (int kk = 0; kk < 32; kk += 4) {
      v2f a[4];
      #pragma unroll
      for (int mt = 0; mt < 4; ++mt)
        a[mt] = *(const v2f*)(&As[(wm * 64 + mt * 16 + l16) * AS_STRIDE + kk + 2 * kh]);
      v2f b[2];
      #pragma unroll
      for (int nt = 0; nt < 2; ++nt) {
        v2f bb;
        bb.x = Bs[(kk + 2 * kh    ) * 128 + wn * 32 + nt * 16 + l16];
        bb.y = Bs[(kk + 2 * kh + 1) * 128 + wn * 32 + nt * 16 + l16];
        b[nt] = bb;
      }
      #pragma unroll
      for (int mt = 0; mt < 4; ++mt)
        #pragma unroll
        for (int nt = 0; nt < 2; ++nt)
          acc[mt][nt] = wmma_f32(a[mt], b[nt], acc[mt][nt]);
    }
  }

  // ---- store C (C/D layout: row = v + 8*kh, col = l16) ----
  #pragma unroll
  for (int mt = 0; mt < 4; ++mt)
    #pragma unroll
    for (int nt = 0; nt < 2; ++nt)
      #pragma unroll
      for (int v = 0; v < 8; ++v) {
        int row = m0 + wm * 64 + mt * 16 + v + 8 * kh;
        int col = n0 + wn * 32 + nt * 16 + l16;
        C[(size_t)row * N + col] = acc[mt][nt][v];
      }
}

// ---------------------------------------------------------------------------
// Flash attention (causal), fp32 WMMA.
// qkv row (stride 3072): head h -> [q(64) k(64) v(64)] at column h*192.
// One workgroup = (64-query block, b*H+h); 4 waves, 16 query rows each.
// K/V 64x64 blocks async-staged into LDS (double-buffered over the jb loop),
// shared by all 4 waves.  Output written (b,h,t,hd)-contiguous, reproducing
// the reference's head-scrambled reshape when read as (b,t,d).
// ---------------------------------------------------------------------------
#define KV_STRIDE 68   // 64+4 pad: rows 272B (16B aligned), banks 4i mod 64
#define P_STRIDE  66   // 64+2 pad: keeps v2f reads 8B aligned

__global__ __launch_bounds__(128)
void attn_flash_wmma_k(const float* __restrict__ qkv, float* __restrict__ vals)
{
  extern __shared__ float smem[];
  const int KV = 64 * KV_STRIDE;
  float* KsB = smem;             // [2][64*KV_STRIDE]
  float* VsB = smem + 2 * KV;    // [2][64*KV_STRIDE]
  float* Pl  = smem + 4 * KV;    // [4][16*P_STRIDE]

  const int tid  = threadIdx.x;
  const int lane = tid & 31;
  const int w    = tid >> 5;    // wave 0..3
  const int l16  = lane & 15;
  const int kh   = lane >> 4;
  const int qb   = blockIdx.x;          // query block (64 rows)
  const int bh   = blockIdx.y;          // b*H + h
  const int b    = bh >> 4;
  const int h    = bh & 15;

  const size_t baseBT = (size_t)b * T_ * D3;
  const int    qrow0  = qb * 64 + w * 16;

  const float* kbase = qkv + baseBT + h * 192 + 64;
  const float* vbase = qkv + baseBT + h * 192 + 128;

  // ---- preload Q strip as A-fragments: qa[kk] covers K = 4*kk .. 4*kk+3 ----
  v2f qa[16];
  {
    const float* qrow = qkv + baseBT + (size_t)(qrow0 + l16) * D3 + h * 192;
    #pragma unroll
    for (int kk = 0; kk < 16; ++kk)
      qa[kk] = *(const v2f*)(qrow + kk * 4 + 2 * kh);
  }

  v8f o[4];
  #pragma unroll
  for (int nt = 0; nt < 4; ++nt)
    #pragma unroll
    for (int v = 0; v < 8; ++v)
      o[nt][v] = 0.0f;

  float mrun[8], lrun[8];
  #pragma unroll
  for (int v = 0; v < 8; ++v) { mrun[v] = -INFINITY; lrun[v] = 0.0f; }

  // ---- async stage of one 64-key K/V block into buffer ib ----
  auto stage_kv = [&](int jb, int ib) {
    const int key0 = jb * 64;
    #pragma unroll
    for (int i = 0; i < 8; ++i) {
      int idx = tid + i * 128;            // 64 rows x 16 float4 slots
      int row = idx >> 4, c4 = idx & 15;
      size_t g = (size_t)(key0 + row) * D3 + c4 * 4;
      async_copy_b128(&KsB[ib * KV + row * KV_STRIDE + c4 * 4], kbase + g);
      async_copy_b128(&VsB[ib * KV + row * KV_STRIDE + c4 * 4], vbase + g);
    }
  };

  stage_kv(0, 0);
  int ib = 0;
  for (int jb = 0; jb <= qb; ++jb, ib ^= 1) {
    wait_asynccnt0();      // buffer-ib K/V in LDS (my copies)
    __syncthreads();       // all copies done; prior readers of ib^1 finished
    if (jb + 1 <= qb) stage_kv(jb + 1, ib ^ 1);

    const float* Ks  = KsB + ib * KV;
    const float* Vs  = VsB + ib * KV;
    const int    key0 = jb * 64;

    // ---- S = Q * K^T from LDS ----
    v8f s[4];
    #pragma unroll
    for (int nt = 0; nt < 4; ++nt)
      #pragma unroll
      for (int v = 0; v < 8; ++v)
        s[nt][v] = 0.0f;

    #pragma unroll
    for (int kk = 0; kk < 16; ++kk) {
      #pragma unroll
      for (int nt = 0; nt < 4; ++nt) {
        v2f kb = *(const v2f*)(&Ks[(nt * 16 + l16) * KV_STRIDE + kk * 4 + 2 * kh]);
        s[nt] = wmma_f32(qa[kk], kb, s[nt]);
      }
    }

    // ---- scale + causal mask + online softmax ----
    const bool diag = (jb == qb);
    #pragma unroll
    for (int v = 0; v < 8; ++v) {
      const int qi = qrow0 + v + 8 * kh;
      #pragma unroll
      for (int nt = 0; nt < 4; ++nt) {
        float val = s[nt][v] * 0.125f;                       // 1/sqrt(64)
        if (diag && (key0 + nt * 16 + l16) > qi) val = -INFINITY;
        s[nt][v] = val;
      }
      // row max across 64 keys: per-register max then 16-lane xor reduction
      float t = fmaxf(fmaxf(s[0][v], s[1][v]), fmaxf(s[2][v], s[3][v]));
      t = fmaxf(t, __shfl_xor(t, 1));
      t = fmaxf(t, __shfl_xor(t, 2));
      t = fmaxf(t, __shfl_xor(t, 4));
      t = fmaxf(t, __shfl_xor(t, 8));

      float mnew  = fmaxf(mrun[v], t);
      float alpha = __expf(mrun[v] - mnew);   // first block: exp(-inf)=0
      mrun[v] = mnew;

      float rs = 0.0f;
      #pragma unroll
      for (int nt = 0; nt < 4; ++nt) {
        float p = __expf(s[nt][v] - mnew);
        s[nt][v] = p;
        rs += p;
      }
      rs += __shfl_xor(rs, 1);
      rs += __shfl_xor(rs, 2);
      rs += __shfl_xor(rs, 4);
      rs += __shfl_xor(rs, 8);
      lrun[v] = lrun[v] * alpha + rs;

      #pragma unroll
      for (int nt = 0; nt < 4; ++nt) o[nt][v] *= alpha;
    }

    // ---- P: C-layout -> A-layout via LDS (each wave owns its strip) ----
    #pragma unroll
    for (int nt = 0; nt < 4; ++nt)
      #pragma unroll
      for (int v = 0; v < 8; ++v)
        Pl[w * (16 * P_STRIDE) + (v + 8 * kh) * P_STRIDE + nt * 16 + l16] = s[nt][v];
    __syncthreads();

    // ---- O += P * V from LDS ----
    #pragma unroll
    for (int kk = 0; kk < 16; ++kk) {
      v2f pa = *(const v2f*)(&Pl[w * (16 * P_STRIDE) + l16 * P_STRIDE + kk * 4 + 2 * kh]);
      #pragma unroll
      for (int nt = 0; nt < 4; ++nt) {
        const float* vp = &Vs[(kk * 4 + 2 * kh) * KV_STRIDE + nt * 16 + l16];
        v2f vb;
        vb.x = vp[0];
        vb.y = vp[KV_STRIDE];
        o[nt] = wmma_f32(pa, vb, o[nt]);
      }
    }
  }

  // ---- normalize and store in (b,h,t,hd) order ----
  #pragma unroll
  for (int nt = 0; nt < 4; ++nt)
    #pragma unroll
    for (int v = 0; v < 8; ++v) {
      int q = qrow0 + v + 8 * kh;
      size_t idx = ((size_t)bh * T_ + q) * HD_ + nt * 16 + l16;
      vals[idx] = o[nt][v] / lrun[v];
    }
}

// ---------------------------------------------------------------------------
extern "C" void kernel_launch(void* const* d_in, const int* in_sizes, int n_in,
                              void* d_out, int out_size, void* d_ws, size_t ws_size,
                              hipStream_t stream) {
  (void)in_sizes; (void)n_in; (void)out_size; (void)ws_size;
  const float* x     = (const float*)d_in[0];   // (B,T,D)
  const float* w_qkv = (const float*)d_in[1];   // (D, 3D)
  const float* w_out = (const float*)d_in[2];   // (D, D)
  float*       out   = (float*)d_out;           // (B,T,D)

  float* qkv  = (float*)d_ws;                       // 4096 x 3072 (48 MB)
  float* vals = qkv + (size_t)BB * T_ * D3;         // 4096 x 1024 (16 MB)

  const size_t gemm_lds = (size_t)(2 * 128 * AS_STRIDE + 2 * 32 * 128) * sizeof(float);
  const size_t attn_lds = (size_t)(4 * 64 * KV_STRIDE + 4 * 16 * P_STRIDE) * sizeof(float);

  // 1) qkv = x @ w_qkv     (4096x1024)*(1024x3072)
  gemm_f32_wmma_k<<<dim3(D3 / 128, (BB * T_) / 128), 256, gemm_lds, stream>>>(
      x, w_qkv, qkv, BB * T_, D3, D_);

  // 2) causal flash attention per (b,h,q-block)
  attn_flash_wmma_k<<<dim3(T_ / 64, BB * H_), 128, attn_lds, stream>>>(qkv, vals);

  // 3) out = vals @ w_out  (4096x1024)*(1024x1024)
  gemm_f32_wmma_k<<<dim3(D_ / 128, (BB * T_) / 128), 256, gemm_lds, stream>>>(
      vals, w_out, out, BB * T_, D_, D_);
}